// MPL_61323543053000
// MI455X (gfx1250) — compile-verified
//
#include <hip/hip_runtime.h>
#include <hip/hip_bf16.h>

typedef __attribute__((ext_vector_type(2))) float v2f;
typedef __attribute__((ext_vector_type(8))) float v8f;

#define IN_FT       256
#define OUT_FT      128
#define ROWS_PER_BLK 64
#define LDS_STRIDE  (IN_FT + 2)   // pad 2 floats: kills LDS bank conflicts on frag reads
#define BN_EPS      1e-5f

// ---------------------------------------------------------------------------
// Kernel 1: h = seq @ W.T  via V_WMMA_F32_16X16X4_F32 (full f32 precision;
// GEMM is memory-bound on MI455X so f32 WMMA is the right precision/speed).
// Block = 256 threads = 8 waves. Block tile: 64 rows x 128 cols.
//   - 64 seq rows staged in LDS (padded stride).
//   - wave w owns col tile [w*16, w*16+16); 4 accumulators reuse each
//     B fragment across 4 row subtiles -> W read once per block from L2.
// ---------------------------------------------------------------------------
__global__ __launch_bounds__(256)
void gcn_gemm_wmma(const float* __restrict__ seq, const float* __restrict__ W,
                   float* __restrict__ h, int N) {
  __shared__ float As[ROWS_PER_BLK * LDS_STRIDE];

  const int rowBase = blockIdx.x * ROWS_PER_BLK;
  const int tid = threadIdx.x;

  // Cooperative stage: 64 rows x 256 floats (clamp rows; keep EXEC full for WMMA).
  for (int i = tid; i < ROWS_PER_BLK * (IN_FT / 4); i += 256) {
    const int r  = i / (IN_FT / 4);
    const int c4 = (i % (IN_FT / 4)) * 4;
    int gr = rowBase + r;
    if (gr >= N) gr = N - 1;                       // clamp instead of branching out
    const float4 v = *(const float4*)(seq + (size_t)gr * IN_FT + c4);
    float* p = &As[r * LDS_STRIDE + c4];
    p[0] = v.x; p[1] = v.y; p[2] = v.z; p[3] = v.w;
  }
  __syncthreads();

  const int lane  = tid & 31;
  const int wave  = tid >> 5;          // 8 waves -> 8 col tiles of 16
  const int l16   = lane & 15;
  const int khalf = (lane >> 4) * 2;   // ISA 32-bit 16x4 A/B layout: lanes 16-31 hold K+2,K+3

  v8f acc[4] = {v8f{}, v8f{}, v8f{}, v8f{}};

  // B fragment source: row (wave*16 + l16) of W, K-contiguous.
  const float* wrow = W + (size_t)(wave * 16 + l16) * IN_FT + khalf;
  // A fragment sources: LDS rows rt*16 + l16.
  const float* a0 = &As[(0 * 16 + l16) * LDS_STRIDE + khalf];
  const float* a1 = &As[(1 * 16 + l16) * LDS_STRIDE + khalf];
  const float* a2 = &As[(2 * 16 + l16) * LDS_STRIDE + khalf];
  const float* a3 = &As[(3 * 16 + l16) * LDS_STRIDE + khalf];

  #pragma unroll 8
  for (int k0 = 0; k0 < IN_FT; k0 += 4) {
    const float2 bw = *(const float2*)(wrow + k0);
    v2f b; b.x = bw.x; b.y = bw.y;

    float2 av; v2f a;

    av = *(const float2*)(a0 + k0); a.x = av.x; a.y = av.y;
    acc[0] = __builtin_amdgcn_wmma_f32_16x16x4_f32(false, a, false, b, (short)0, acc[0], false, false);
    av = *(const float2*)(a1 + k0); a.x = av.x; a.y = av.y;
    acc[1] = __builtin_amdgcn_wmma_f32_16x16x4_f32(false, a, false, b, (short)0, acc[1], false, false);
    av = *(const float2*)(a2 + k0); a.x = av.x; a.y = av.y;
    acc[2] = __builtin_amdgcn_wmma_f32_16x16x4_f32(false, a, false, b, (short)0, acc[2], false, false);
    av = *(const float2*)(a3 + k0); a.x = av.x; a.y = av.y;
    acc[3] = __builtin_amdgcn_wmma_f32_16x16x4_f32(false, a, false, b, (short)0, acc[3], false, false);
  }

  // C/D layout: VGPR r -> M = r (lanes 0-15) / M = r+8 (lanes 16-31), N = lane&15.
  const int col  = wave * 16 + l16;
  const int rOff = (lane >> 4) * 8;
  #pragma unroll
  for (int rt = 0; rt < 4; ++rt) {
    #pragma unroll
    for (int r = 0; r < 8; ++r) {
      const int row = rowBase + rt * 16 + rOff + r;
      if (row < N) h[(size_t)row * OUT_FT + col] = acc[rt][r];
    }
  }
}

// ---------------------------------------------------------------------------
// Kernel 2: edge scatter: agg[dst[e]] += h[src[e]] * w[e].
// 32 lanes per edge, float4 per lane (128 features). agg (51 MB) is
// L2-resident on MI455X (192 MB L2) so the f32 atomics resolve in L2.
// ---------------------------------------------------------------------------
__global__ __launch_bounds__(256)
void gcn_edge_scatter(const float* __restrict__ h, const int* __restrict__ src,
                      const int* __restrict__ dst, const float* __restrict__ ew,
                      float* __restrict__ agg, int E) {
  const long long t = (long long)blockIdx.x * blockDim.x + threadIdx.x;
  const int e = (int)(t >> 5);
  if (e >= E) return;
  const int lane = (int)(t & 31);

  const int   s = src[e];
  const int   d = dst[e];
  const float w = ew[e];

  const float4 v = ((const float4*)(h + (size_t)s * OUT_FT))[lane];
  float* ap = agg + (size_t)d * OUT_FT + lane * 4;
  atomicAdd(ap + 0, v.x * w);
  atomicAdd(ap + 1, v.y * w);
  atomicAdd(ap + 2, v.z * w);
  atomicAdd(ap + 3, v.w * w);
}

// ---------------------------------------------------------------------------
// Kernel 3: per-feature sum and sum-of-squares (coalesced: 128 threads = 1 row).
// ---------------------------------------------------------------------------
__global__ __launch_bounds__(128)
void gcn_bn_stats(const float* __restrict__ agg, float* __restrict__ sums, int N) {
  const int f = threadIdx.x;            // feature 0..127
  float s = 0.f, sq = 0.f;
  for (int r = blockIdx.x; r < N; r += gridDim.x) {
    const float v = agg[(size_t)r * OUT_FT + f];
    s += v;
    sq += v * v;
  }
  atomicAdd(&sums[f], s);
  atomicAdd(&sums[OUT_FT + f], sq);
}

// ---------------------------------------------------------------------------
// Kernel 4: out = tanh(gamma*(agg-mean)*rsqrt(var+eps) + beta + bias)
// ---------------------------------------------------------------------------
__global__ __launch_bounds__(256)
void gcn_bn_tanh(const float* __restrict__ agg, const float* __restrict__ sums,
                 const float* __restrict__ gamma, const float* __restrict__ beta,
                 const float* __restrict__ bias, float* __restrict__ out, int N) {
  const size_t i = (size_t)blockIdx.x * blockDim.x + threadIdx.x;
  const size_t total = (size_t)N * OUT_FT;
  if (i >= total) return;
  const int f = (int)(i & (OUT_FT - 1));
  const float invN = 1.0f / (float)N;
  const float mean = sums[f] * invN;
  const float var  = sums[OUT_FT + f] * invN - mean * mean;
  const float inv  = rsqrtf(var + BN_EPS);
  const float v = (agg[i] - mean) * inv * gamma[f] + beta[f] + bias[f];
  out[i] = tanhf(v);
}

// ---------------------------------------------------------------------------
// Launch. Inputs (setup_inputs order):
//   0 seq [N*256] f32, 1 W [128*256] f32, 2 gamma[128], 3 beta[128],
//   4 bias[128], 5 edge_w[E] f32, 6 src[E] i32, 7 dst[E] i32.
// Workspace: h [N*128] | agg [N*128] | sums [256]  (f32)
// ---------------------------------------------------------------------------
extern "C" void kernel_launch(void* const* d_in, const int* in_sizes, int n_in,
                              void* d_out, int out_size, void* d_ws, size_t ws_size,
                              hipStream_t stream) {
  const float* seq   = (const float*)d_in[0];
  const float* W     = (const float*)d_in[1];
  const float* gamma = (const float*)d_in[2];
  const float* beta  = (const float*)d_in[3];
  const float* bias  = (const float*)d_in[4];
  const float* ew    = (const float*)d_in[5];
  const int*   src   = (const int*)d_in[6];
  const int*   dst   = (const int*)d_in[7];

  const int N = in_sizes[0] / IN_FT;
  const int E = in_sizes[5];

  float* h    = (float*)d_ws;
  float* agg  = h + (size_t)N * OUT_FT;
  float* sums = agg + (size_t)N * OUT_FT;

  // zero agg + sums (contiguous), capture-safe memset node
  hipMemsetAsync(agg, 0, ((size_t)N * OUT_FT + 2 * OUT_FT) * sizeof(float), stream);

  // 1) GEMM (WMMA f32)
  const int gemmBlocks = (N + ROWS_PER_BLK - 1) / ROWS_PER_BLK;
  gcn_gemm_wmma<<<gemmBlocks, 256, 0, stream>>>(seq, W, h, N);

  // 2) edge scatter-add (32 lanes/edge)
  const long long threads = (long long)E * 32;
  const int scatterBlocks = (int)((threads + 255) / 256);
  gcn_edge_scatter<<<scatterBlocks, 256, 0, stream>>>(h, src, dst, ew, agg, E);

  // 3) BN stats
  gcn_bn_stats<<<512, 128, 0, stream>>>(agg, sums, N);

  // 4) normalize + tanh
  const size_t total = (size_t)N * OUT_FT;
  const int outBlocks = (int)((total + 255) / 256);
  gcn_bn_tanh<<<outBlocks, 256, 0, stream>>>(agg, sums, gamma, beta, bias,
                                             (float*)d_out, N);
}